// EGAD_67156108640611
// MI455X (gfx1250) — compile-verified
//
#include <hip/hip_runtime.h>
#include <hip/hip_bf16.h>

typedef float v2f __attribute__((ext_vector_type(2)));
typedef float v4f __attribute__((ext_vector_type(4)));
typedef float v8f __attribute__((ext_vector_type(8)));

#define TILE_EDGES   128     // edges per block (8 waves x 16)
#define WAVE_EDGES   16
#define K_DIM        72      // 4 (src) + 4 (dst) + 64 (edge feat)
#define LDS_STRIDE   76      // floats per LDS row: 304 B, 16B-aligned, good banking
#define NSTEPS       18      // 72 / 4

__global__ __launch_bounds__(256)
void egad_edge_gemv_wmma(const float* __restrict__ node_embed,
                         const float* __restrict__ edge_feat,
                         const float* __restrict__ W,      // [72][2] row-major
                         const float* __restrict__ bvec,   // [2]
                         const int*   __restrict__ edges,  // [NUM_EDGES][2]
                         const int*   __restrict__ edge_indices, // [batch]
                         float* __restrict__ out,          // [batch][2]
                         int batch)
{
    __shared__ float tile[TILE_EDGES * LDS_STRIDE];   // 38,912 B
    __shared__ int   eidx_s[TILE_EDGES];              // staged gather indices

    const int tid        = threadIdx.x;
    const int lane       = tid & 31;
    const int wave       = tid >> 5;
    const int block_base = blockIdx.x * TILE_EDGES;

    // ------------------------------------------------------------------
    // Stage edge_indices for this tile into LDS (coalesced, once).
    // ------------------------------------------------------------------
    if (tid < TILE_EDGES) {
        int e = block_base + tid;
        if (e > batch - 1) e = batch - 1;             // clamp tail (stores guarded later)
        eidx_s[tid] = edge_indices[e];
    }

    // ------------------------------------------------------------------
    // Build B fragments for V_WMMA_F32_16X16X4_F32 (same for every tile).
    // B is 4x16 (KxN), f32, 2 VGPRs:
    //   VGPR0: lanes 0-15 row K=0, lanes 16-31 row K=2
    //   VGPR1: lanes 0-15 row K=1, lanes 16-31 row K=3
    // W is 72x2; pad N to 16 with zeros.  Fully branchless: every lane
    // loads a valid (clamped) address, dead lanes select 0 via v_cndmask.
    // ------------------------------------------------------------------
    const int  col  = lane & 15;          // N index this lane owns
    const int  colc = col & 1;            // clamped column for the load
    const bool live = (col < 2);
    const int  koff = (lane >> 4) << 1;   // 0 for lanes 0-15, 2 for lanes 16-31

    const float btmp = bvec[colc];        // unconditional load, then select
    const float bias = live ? btmp : 0.0f;

    v2f bfrag[NSTEPS];
#pragma unroll
    for (int s = 0; s < NSTEPS; ++s) {
        const int k = 4 * s + koff;       // k, k+1 < 72 always
        const float bx = W[k * 2 + colc];
        const float by = W[(k + 1) * 2 + colc];
        bfrag[s].x = live ? bx : 0.0f;
        bfrag[s].y = live ? by : 0.0f;
    }

    __syncthreads();

    // ------------------------------------------------------------------
    // Gather phase: stage [src(4) | dst(4) | ef(64)] rows into LDS.
    // ------------------------------------------------------------------
    // Node embeddings: 2 threads per edge, one b128 load each.
    {
        const int e_local = tid >> 1;                 // 0..127
        const int which   = tid & 1;                  // 0=src, 1=dst
        const int eidx = eidx_s[e_local];
        const int node = edges[(size_t)eidx * 2 + which];
        const v4f nv = *(const v4f*)(node_embed + (size_t)node * 4);
        *(v4f*)(tile + e_local * LDS_STRIDE + which * 4) = nv;
    }

    // Edge features: 16 threads per edge, each edge's 256B row read as 16
    // contiguous b128s by 16 adjacent lanes (fully coalesced).
#pragma unroll
    for (int i = 0; i < 8; ++i) {
        const int idx     = i * 256 + tid;            // 0..2047
        const int e_local = idx >> 4;                 // 0..127
        const int chunk   = idx & 15;                 // float4 chunk within row
        const int eidx = eidx_s[e_local];
        const v4f fv = *(const v4f*)(edge_feat + (size_t)eidx * 64 + chunk * 4);
        *(v4f*)(tile + e_local * LDS_STRIDE + 8 + chunk * 4) = fv;
    }

    __syncthreads();

    // ------------------------------------------------------------------
    // Compute: each wave owns 16 edges; 18 x V_WMMA_F32_16X16X4_F32.
    // A is 16x4 (MxK), f32, 2 VGPRs:
    //   lanes 0-15 hold (M=lane, K=4s..4s+1), lanes 16-31 (M=lane-16, K=4s+2..4s+3)
    // ------------------------------------------------------------------
    const int row = lane & 15;
    const float* arow = tile + (wave * WAVE_EDGES + row) * LDS_STRIDE + koff;

    v8f acc = {};
#pragma unroll
    for (int s = 0; s < NSTEPS; ++s) {
        const v2f a = *(const v2f*)(arow + 4 * s);
        acc = __builtin_amdgcn_wmma_f32_16x16x4_f32(
            /*neg_a=*/false, a,
            /*neg_b=*/false, bfrag[s],
            /*c_mod=*/(short)0, acc,
            /*reuse_a=*/false, /*reuse_b=*/false);
    }

    // ------------------------------------------------------------------
    // Write-out: D is 16x16 f32; VGPR r holds (M=r, N=lane) for lanes 0-15
    // and (M=8+r, N=lane-16) for lanes 16-31. Only N=0,1 are live.
    // ------------------------------------------------------------------
    if (live) {
        const int mbase = (lane >> 4) * 8;
        const int e0 = block_base + wave * WAVE_EDGES + mbase;
#pragma unroll
        for (int r = 0; r < 8; ++r) {
            const int e = e0 + r;
            if (e < batch) {
                out[(size_t)e * 2 + col] = acc[r] + bias;
            }
        }
    }
}

extern "C" void kernel_launch(void* const* d_in, const int* in_sizes, int n_in,
                              void* d_out, int out_size, void* d_ws, size_t ws_size,
                              hipStream_t stream) {
    (void)n_in; (void)out_size; (void)d_ws; (void)ws_size;
    const float* node_embed   = (const float*)d_in[0];
    const float* edge_feat    = (const float*)d_in[1];
    const float* W            = (const float*)d_in[2];
    const float* bvec         = (const float*)d_in[3];
    const int*   edges        = (const int*)d_in[4];
    const int*   edge_indices = (const int*)d_in[5];
    float* out = (float*)d_out;

    const int batch = in_sizes[5];
    const int grid  = (batch + TILE_EDGES - 1) / TILE_EDGES;

    hipLaunchKernelGGL(egad_edge_gemv_wmma, dim3(grid), dim3(256), 0, stream,
                       node_embed, edge_feat, W, bvec, edges, edge_indices,
                       out, batch);
}